// ResidualQuantizer_89850715833213
// MI455X (gfx1250) — compile-verified
//
#include <hip/hip_runtime.h>

// Residual quantizer for MI455X (gfx1250, wave32, WMMA).
// N=65536 tokens, D=256, L=4 levels, K=2048 codes/level.
//
// Dominant cost: 4 x (N*K*D*2) = 275 GFLOP of fp32 matmul -> v_wmma_f32_16x16x4_f32.
// Codebooks (8 MB) live in the 192 MB L2; outputs (~605 MB) cost ~26 us at 23.3 TB/s.
// The distance matrix is never materialized: argmin is fused into the GEMM.

#define DIM   256
#define KCB   2048
#define LVL   4
#define KSTEPS (DIM / 4)     // 64 wmma k-steps per 16x16 score tile
#define CTILES (KCB / 16)    // 128 column tiles of 16 codes

typedef float v8f  __attribute__((ext_vector_type(8)));
typedef float v2f  __attribute__((ext_vector_type(2)));
typedef float v4f  __attribute__((ext_vector_type(4)));

// ---------------------------------------------------------------------------
// Kernel 1 (per level): fused  S = R*W^T  (wmma f32 16x16x4)  +  argmin of
// (||w||^2 - 2 r.w).  One wave per 16-token tile; A-panel held in VGPRs,
// W streamed from L2, ||w||^2 folded into the B-load stream.
// Four independent WMMA accumulator chains hide matrix-pipe latency even at
// low occupancy.
// ---------------------------------------------------------------------------
__global__ __launch_bounds__(256) void rq_argmin_kernel(
    const float* __restrict__ R, int rStride,     // residual rows (stride in floats)
    const float* __restrict__ W,                  // [K, D] this level's codebook
    float* __restrict__ sids,                     // [N, L] written as float
    int level)
{
  const int lane = threadIdx.x & 31;
  const int wave = threadIdx.x >> 5;
  const int tile = blockIdx.x * (blockDim.x >> 5) + wave;
  const int row0 = tile << 4;            // first token of this wave's tile
  const int nl   = lane & 15;
  const int half = lane >> 4;            // 0: K%4 in {0,1}, 1: K%4 in {2,3}

  // A-panel (16 tokens x 256) resident in registers, in WMMA f32 16x16x4
  // layout: lanes 0-15 -> M=lane, K = 4s+{0,1}; lanes 16-31 -> M=lane-16,
  // K = 4s+{2,3}.
  float a0[KSTEPS], a1[KSTEPS];
  {
    const float* rrow = R + (size_t)(row0 + nl) * rStride + half * 2;
    #pragma unroll
    for (int s = 0; s < KSTEPS; ++s) {
      a0[s] = rrow[4 * s];
      a1[s] = rrow[4 * s + 1];
    }
  }

  float bestVal[8];
  int   bestIdx[8];
  #pragma unroll
  for (int v = 0; v < 8; ++v) { bestVal[v] = 3.402823466e+38f; bestIdx[v] = 0x7fffffff; }

  for (int ct = 0; ct < CTILES; ++ct) {
    const int code = (ct << 4) + nl;     // this lane's code column
    const float* wrow = W + (size_t)code * DIM + half * 2;

    // 4 independent accumulator chains -> 4-way ILP on the matrix pipe.
    v8f acc[4];
    #pragma unroll
    for (int c = 0; c < 4; ++c) acc[c] = (v8f){0.f,0.f,0.f,0.f,0.f,0.f,0.f,0.f};
    float nrm0 = 0.f, nrm1 = 0.f;        // partial ||w||^2 over this lane's K slices

    #pragma unroll
    for (int s = 0; s < KSTEPS; ++s) {
      v2f a; a.x = a0[s];        a.y = a1[s];
      v2f b; b.x = wrow[4 * s];  b.y = wrow[4 * s + 1];
      nrm0 = __builtin_fmaf(b.x, b.x, nrm0);
      nrm1 = __builtin_fmaf(b.y, b.y, nrm1);
      acc[s & 3] = __builtin_amdgcn_wmma_f32_16x16x4_f32(
                       false, a, false, b, (short)0, acc[s & 3], false, false);
    }

    // join the two K-halves -> full ||w||^2 for this lane's code
    float nrm = nrm0 + nrm1;
    nrm += __shfl_xor(nrm, 16, 32);

    #pragma unroll
    for (int v = 0; v < 8; ++v) {
      float s03 = (acc[0][v] + acc[1][v]) + (acc[2][v] + acc[3][v]);
      // ||r||^2 is constant per row -> argmin of (||w||^2 - 2 r.w)
      float dist = __builtin_fmaf(-2.f, s03, nrm);
      if (dist < bestVal[v] || (dist == bestVal[v] && code < bestIdx[v])) {
        bestVal[v] = dist; bestIdx[v] = code;
      }
    }
  }

  // Cross-lane argmin inside each 16-lane half (all 16 lanes of a half share
  // the same token per C-slot: token = v + 8*half). First-index tie-break.
  #pragma unroll
  for (int v = 0; v < 8; ++v) {
    float val = bestVal[v];
    int   idx = bestIdx[v];
    #pragma unroll
    for (int m = 8; m >= 1; m >>= 1) {
      float ov = __shfl_xor(val, m, 32);
      int   oi = __shfl_xor(idx, m, 32);
      if (ov < val || (ov == val && oi < idx)) { val = ov; idx = oi; }
    }
    if (nl == 0) {
      int token = row0 + v + (half << 3);
      sids[(size_t)token * LVL + level] = (float)idx;
    }
  }
}

// ---------------------------------------------------------------------------
// Kernel 2 (per level): gather q = W[idx], emit residuals/quantized, write the
// next level's residual in-place into the residuals output region, accumulate
// z_q. Pure bandwidth kernel, float4 everywhere.
// ---------------------------------------------------------------------------
__global__ __launch_bounds__(256) void rq_update_kernel(
    const float* __restrict__ Rin, int rStride,
    const float* __restrict__ W,
    const float* __restrict__ sids,
    float* __restrict__ zq,          // [N, D]
    float* __restrict__ residuals,   // [N, L, D]
    float* __restrict__ quantized,   // [N, L, D]
    const int* __restrict__ activePtr,
    int level)
{
  int e     = blockIdx.x * blockDim.x + threadIdx.x;  // one float4 each
  int token = e >> 6;                                 // D/4 = 64 float4 per token
  int d     = (e & 63) << 2;

  int active = *activePtr;
  active = active < 1 ? 1 : (active > LVL ? LVL : active);

  int idx = (int)sids[(size_t)token * LVL + level];

  v4f q = *(const v4f*)(W + (size_t)idx * DIM + d);
  v4f r = *(const v4f*)(Rin + (size_t)token * rStride + d);

  size_t base = ((size_t)token * LVL + level) * DIM + d;
  *(v4f*)(residuals + base) = r;
  *(v4f*)(quantized + base) = q;
  if (level + 1 < LVL)
    *(v4f*)(residuals + base + DIM) = r - q;   // next level's input residual

  float msk = (level < active) ? 1.f : 0.f;
  v4f qz = { q.x * msk, q.y * msk, q.z * msk, q.w * msk };
  float* z = zq + (size_t)token * DIM + d;
  if (level == 0) {
    *(v4f*)z = qz;
  } else {
    v4f cur = *(v4f*)z;
    cur.x += qz.x; cur.y += qz.y; cur.z += qz.z; cur.w += qz.w;
    *(v4f*)z = cur;
  }
}

__global__ void rq_mask_kernel(float* __restrict__ amask,
                               const int* __restrict__ activePtr)
{
  int active = *activePtr;
  active = active < 1 ? 1 : (active > LVL ? LVL : active);
  int t = threadIdx.x;
  if (t < LVL) amask[t] = (t < active) ? 1.f : 0.f;
}

// ---------------------------------------------------------------------------
extern "C" void kernel_launch(void* const* d_in, const int* in_sizes, int n_in,
                              void* d_out, int out_size, void* d_ws, size_t ws_size,
                              hipStream_t stream) {
  (void)n_in; (void)out_size; (void)d_ws; (void)ws_size;

  const float* h   = (const float*)d_in[0];   // [N, D]
  const float* cb  = (const float*)d_in[1];   // [L, K, D]
  const int*   act = (const int*)  d_in[2];   // scalar

  const int N = in_sizes[0] / DIM;            // 65536

  float* out       = (float*)d_out;
  float* zq        = out;                                  // [N, D]
  float* sids      = zq        + (size_t)N * DIM;          // [N, L]
  float* residuals = sids      + (size_t)N * LVL;          // [N, L, D]
  float* quantized = residuals + (size_t)N * LVL * DIM;    // [N, L, D]
  float* amask     = quantized + (size_t)N * LVL * DIM;    // [L]

  rq_mask_kernel<<<1, LVL, 0, stream>>>(amask, act);

  const int tiles        = N / 16;                 // waves for argmin
  const int argminBlocks = tiles / 8;              // 8 waves (256 thr) per block
  const int updThreads   = N * (DIM / 4);          // one float4 per thread
  const int updBlocks    = updThreads / 256;

  for (int l = 0; l < LVL; ++l) {
    const float* W = cb + (size_t)l * KCB * DIM;
    const float* R = (l == 0) ? h : residuals + (size_t)l * DIM;
    int rStride    = (l == 0) ? DIM : LVL * DIM;

    rq_argmin_kernel<<<argminBlocks, 256, 0, stream>>>(R, rStride, W, sids, l);
    rq_update_kernel<<<updBlocks, 256, 0, stream>>>(R, rStride, W, sids,
                                                    zq, residuals, quantized,
                                                    act, l);
  }
}